// AttMaxPool2D_35399120454254
// MI455X (gfx1250) — compile-verified
//
#include <hip/hip_runtime.h>

// AttMaxPool2D: softmax-weighted 2x2 pooling over x[32,128,128,128] f32.
// out[b,wo,ho,c] = sum_i v_i * softmax_i(v*T), v = 2x2 patch values.
//
// Memory-bound (320 MB traffic @ 23.3 TB/s ~ 13.7us), zero data reuse:
//  - per block: stage 2 input rows x 64 cols x 128 ch (64 KB, contiguous per
//    row) into LDS with gfx1250 async global->LDS B128 copies (ASYNCcnt),
//    non-temporal hint (input stream 256 MB > 192 MB L2, read exactly once),
//  - s_wait_asynccnt 0 + workgroup barrier,
//  - compute softmax pooling from LDS (ds_load_b128), non-temporal B128 store.

#define BLOCK 256
#define LOG2E 1.44269504088896340736f

typedef __attribute__((ext_vector_type(4))) float v4f;  // native vec for NT store

__device__ __forceinline__ float attn4(float a, float b, float c, float d,
                                       float Tk2) {
  // scaled-by-log2e softmax domain; subtracting max in this domain is the
  // same softmax as exp((v*T) - max(v*T)).
  float sa = a * Tk2, sb = b * Tk2, sc = c * Tk2, sd = d * Tk2;
  float m = fmaxf(fmaxf(sa, sb), fmaxf(sc, sd));
  float ea = __builtin_amdgcn_exp2f(sa - m);
  float eb = __builtin_amdgcn_exp2f(sb - m);
  float ec = __builtin_amdgcn_exp2f(sc - m);
  float ed = __builtin_amdgcn_exp2f(sd - m);
  float num = fmaf(a, ea, fmaf(b, eb, fmaf(c, ec, d * ed)));
  float den = (ea + eb) + (ec + ed);
  return num * __builtin_amdgcn_rcpf(den);
}

__global__ __launch_bounds__(BLOCK) void
AttMaxPool2D_35399120454254_kernel(const float* __restrict__ x,
                                   const float* __restrict__ temperature,
                                   float* __restrict__ out) {
  // Tile: input rows w = {2*wo, 2*wo+1}, input cols [h0, h0+64), all 128 ch.
  // smem[r * 8192 + hl * 128 + c], r in {0,1}, hl in [0,64), c in [0,128).
  __shared__ float smem[2 * 64 * 128];  // 16384 floats = 64 KB

  const int tid  = threadIdx.x;
  const int bid  = blockIdx.x;
  const int half = bid & 1;          // which half of the output row
  const int wo   = (bid >> 1) & 63;  // output row
  const int b    = bid >> 7;         // batch

  const int   h0  = half * 64;       // input col start
  const float Tk2 = temperature[0] * LOG2E;

  // ---- stage: 2 rows x (64 cols * 128 ch) = 2 x 32 KB contiguous runs ----
  // 2048 float4 copies per row, 8 per thread per row, 16 async ops total.
  // TH_LOAD_NT: pure streaming input, read once, larger than L2.
  #pragma unroll
  for (int r = 0; r < 2; ++r) {
    const float* rowp =
        x + ((size_t)(b * 128 + 2 * wo + r) * 128 + h0) * 128;
    unsigned lbase = (unsigned)(size_t)&smem[r * 8192];
    #pragma unroll
    for (int k = 0; k < 8; ++k) {
      int i = tid + k * BLOCK;                 // float4 index in this row tile
      unsigned     laddr = lbase + (unsigned)i * 16u;
      const float* gaddr = rowp + (size_t)i * 4;
      asm volatile("global_load_async_to_lds_b128 %0, %1, off th:TH_LOAD_NT"
                   :: "v"(laddr), "v"(gaddr)
                   : "memory");
    }
  }
  asm volatile("s_wait_asynccnt 0" ::: "memory");
  __syncthreads();

  // ---- compute: 32 output pixels x 32 float4 = 1024 float4; 4 per thread ----
  float* obase = out + ((size_t)(b * 64 + wo) * 64 + half * 32) * 128;
  #pragma unroll
  for (int k = 0; k < 4; ++k) {
    int q  = tid + k * BLOCK;  // [0, 1024)
    int hl = q >> 5;           // output pixel within tile [0, 32)
    int c4 = q & 31;           // float4 channel group [0, 32)

    const v4f v0 = *(const v4f*)&smem[(2 * hl)     * 128 + c4 * 4];
    const v4f v1 = *(const v4f*)&smem[(2 * hl + 1) * 128 + c4 * 4];
    const v4f v2 = *(const v4f*)&smem[8192 + (2 * hl)     * 128 + c4 * 4];
    const v4f v3 = *(const v4f*)&smem[8192 + (2 * hl + 1) * 128 + c4 * 4];

    v4f o;
    o.x = attn4(v0.x, v1.x, v2.x, v3.x, Tk2);
    o.y = attn4(v0.y, v1.y, v2.y, v3.y, Tk2);
    o.z = attn4(v0.z, v1.z, v2.z, v3.z, Tk2);
    o.w = attn4(v0.w, v1.w, v2.w, v3.w, Tk2);

    // Output is also write-once streaming: non-temporal B128 store.
    __builtin_nontemporal_store(o, (v4f*)&obase[hl * 128 + c4 * 4]);
  }
}

extern "C" void kernel_launch(void* const* d_in, const int* in_sizes, int n_in,
                              void* d_out, int out_size, void* d_ws,
                              size_t ws_size, hipStream_t stream) {
  (void)in_sizes; (void)n_in; (void)d_ws; (void)ws_size; (void)out_size;
  const float* x    = (const float*)d_in[0];
  const float* temp = (const float*)d_in[1];
  float*       out  = (float*)d_out;

  // grid: B(32) * Wo(64) * 2 halves = 4096 blocks of 256 threads.
  AttMaxPool2D_35399120454254_kernel<<<dim3(32 * 64 * 2), dim3(BLOCK), 0,
                                       stream>>>(x, temp, out);
}